// AdvancedDeltaMLModel_85873576116383
// MI455X (gfx1250) — compile-verified
//
#include <hip/hip_runtime.h>
#include <math.h>

#define HID 64
#define EDGE_D 16
#define GLOB_D 16
#define N_LAYERS 3
#define EPS 1e-5f

typedef __attribute__((ext_vector_type(2))) float v2f;
typedef __attribute__((ext_vector_type(8))) float v8f;

// D = A(16x4) * B(4x16) + C   -- exact f32 accumulation (matches reference numerics)
__device__ __forceinline__ v8f wmma_f32_16x16x4(v2f a, v2f b, v8f c) {
  return __builtin_amdgcn_wmma_f32_16x16x4_f32(
      /*neg_a=*/false, a, /*neg_b=*/false, b,
      /*c_mod=*/(short)0, c, /*reuse_a=*/false, /*reuse_b=*/false);
}

// ---------------------------------------------------------------- utilities
__global__ void zero_f32(float* __restrict__ p, long n) {
  long i = (long)blockIdx.x * blockDim.x + threadIdx.x;
  if (i < n) p[i] = 0.0f;
}

__global__ void copy_f32(float* __restrict__ d, const float* __restrict__ s, long n) {
  long i = (long)blockIdx.x * blockDim.x + threadIdx.x;
  if (i < n) d[i] = s[i];
}

// ------------------------------------------------- layer-invariant edge sums
// EAsum[n][0:16] = sum over edges with dst==n of edge_attr[e], deg[n] = in-degree
__global__ void edge_presum(const float* __restrict__ ea, const int* __restrict__ dst,
                            float* __restrict__ EAsum, float* __restrict__ deg, long E) {
  long t = (long)blockIdx.x * blockDim.x + threadIdx.x;
  if (t >= E * EDGE_D) return;
  long e = t >> 4;
  int c = (int)(t & 15);
  int d = dst[e];
  atomicAdd(&EAsum[(long)d * EDGE_D + c], ea[e * EDGE_D + c]);
  if (c == 0) atomicAdd(&deg[d], 1.0f);
}

// ------------------------------------------------- per-layer weight folding
// M1 = W1 @ W2[:64]   (64x64),  bm = b1 @ W2[:64] + b2  (64),
// gu[g] = global_feature[g] @ Wu[64:] + bu  (G x 64)
__global__ void fold_weights(const float* __restrict__ W1, const float* __restrict__ b1,
                             const float* __restrict__ W2, const float* __restrict__ b2,
                             const float* __restrict__ Wu, const float* __restrict__ bu,
                             const float* __restrict__ gf, int G,
                             float* __restrict__ M1, float* __restrict__ bm,
                             float* __restrict__ gu) {
  int t = threadIdx.x;
  for (int idx = t; idx < HID * HID; idx += blockDim.x) {
    int r = idx >> 6, c = idx & 63;
    float acc = 0.f;
    for (int k = 0; k < HID; ++k) acc += W1[r * HID + k] * W2[k * HID + c];
    M1[idx] = acc;
  }
  for (int c = t; c < HID; c += blockDim.x) {
    float acc = b2[c];
    for (int k = 0; k < HID; ++k) acc += b1[k] * W2[k * HID + c];
    bm[c] = acc;
  }
  for (int idx = t; idx < G * HID; idx += blockDim.x) {
    int g = idx >> 6, c = idx & 63;
    float acc = bu[c];
    for (int j = 0; j < GLOB_D; ++j) acc += gf[g * GLOB_D + j] * Wu[(HID + j) * HID + c];
    gu[idx] = acc;
  }
}

// ------------------------------------------------- WMMA GEMM: OUT = IN @ W (N x 64 @ 64 x 64)
// 128 threads = 4 waves; each wave owns a 16x16 output tile of a 16x64 strip.
__global__ __launch_bounds__(128) void gemm_n64_wmma(const float* __restrict__ IN,
                                                     const float* __restrict__ W,
                                                     float* __restrict__ OUT, int N) {
  __shared__ float sW[HID * HID];
  int t = threadIdx.x;
  for (int i = t; i < HID * HID; i += 128) sW[i] = W[i];
  __syncthreads();
  int wave = t >> 5, lane = t & 31;
  int m = lane & 15, half = lane >> 4;
  int row0 = blockIdx.x * 16;
  int col0 = wave * 16;
  int arow = row0 + m;
  if (arow > N - 1) arow = N - 1;
  const float* A = IN + (long)arow * HID;
  v8f acc = {};
  for (int kk = 0; kk < HID; kk += 4) {
    int ka = kk + half * 2;                  // lane halves carry K and K+2 (ISA 16x4 layout)
    v2f a, b;
    a[0] = A[ka];
    a[1] = A[ka + 1];
    b[0] = sW[ka * HID + col0 + m];
    b[1] = sW[(ka + 1) * HID + col0 + m];
    acc = wmma_f32_16x16x4(a, b, acc);
  }
  for (int j = 0; j < 8; ++j) {             // C/D: VGPR j -> rows j (lanes 0-15) / j+8 (16-31)
    int r = row0 + j + half * 8;
    if (r < N) OUT[(long)r * HID + col0 + m] = acc[j];
  }
}

// ------------------------------------------------- aggr init with folded edge terms
// aggr[n][c] = deg[n]*bm[c] + EAsum[n] . W2e[:,c]
__global__ void init_aggr(const float* __restrict__ EAsum, const float* __restrict__ deg,
                          const float* __restrict__ W2e, const float* __restrict__ bm,
                          float* __restrict__ aggr, int N) {
  long t = (long)blockIdx.x * blockDim.x + threadIdx.x;
  if (t >= (long)N * HID) return;
  long n = t >> 6;
  int c = (int)(t & 63);
  float acc = deg[n] * bm[c];
  const float* e = EAsum + n * EDGE_D;
  for (int j = 0; j < EDGE_D; ++j) acc += e[j] * W2e[j * HID + c];
  aggr[t] = acc;
}

// ------------------------------------------------- edge scatter: aggr[dst] += p[src]
__global__ void scatter_edges(const float* __restrict__ p, const int* __restrict__ src,
                              const int* __restrict__ dst, float* __restrict__ aggr, long E) {
  long t = (long)blockIdx.x * blockDim.x + threadIdx.x;
  if (t >= E * HID) return;
  long e = t >> 6;
  int c = (int)(t & 63);
  int s = src[e], d = dst[e];
  atomicAdd(&aggr[(long)d * HID + c], p[(long)s * HID + c]);
}

// ------------------------------------------------- WMMA GEMM + gu add + LayerNorm -> x_cur
__global__ __launch_bounds__(128) void update_ln_wmma(
    const float* __restrict__ AG, const float* __restrict__ Wut,
    const float* __restrict__ gu, const int* __restrict__ batch,
    const float* __restrict__ lnw, const float* __restrict__ lnb,
    float* __restrict__ XO, int N) {
  __shared__ float sW[HID * HID];
  __shared__ float sh[16 * HID];
  __shared__ float smu[16];
  __shared__ float srs[16];
  int t = threadIdx.x;
  for (int i = t; i < HID * HID; i += 128) sW[i] = Wut[i];
  __syncthreads();
  int wave = t >> 5, lane = t & 31;
  int m = lane & 15, half = lane >> 4;
  int row0 = blockIdx.x * 16;
  int col0 = wave * 16;
  int arow = row0 + m;
  if (arow > N - 1) arow = N - 1;
  const float* A = AG + (long)arow * HID;
  v8f acc = {};
  for (int kk = 0; kk < HID; kk += 4) {
    int ka = kk + half * 2;
    v2f a, b;
    a[0] = A[ka];
    a[1] = A[ka + 1];
    b[0] = sW[ka * HID + col0 + m];
    b[1] = sW[(ka + 1) * HID + col0 + m];
    acc = wmma_f32_16x16x4(a, b, acc);
  }
  for (int j = 0; j < 8; ++j) {
    int r = j + half * 8;
    int row = row0 + r;
    if (row > N - 1) row = N - 1;
    int g = batch[row];
    sh[r * HID + col0 + m] = acc[j] + gu[(long)g * HID + col0 + m];
  }
  __syncthreads();
  if (t < 16) {
    const float* hr = sh + t * HID;
    float mu = 0.f;
    for (int c = 0; c < HID; ++c) mu += hr[c];
    mu *= (1.0f / HID);
    float var = 0.f;
    for (int c = 0; c < HID; ++c) { float d = hr[c] - mu; var += d * d; }
    var *= (1.0f / HID);
    smu[t] = mu;
    srs[t] = rsqrtf(var + EPS);
  }
  __syncthreads();
  for (int i = t; i < 16 * HID; i += 128) {
    int r = i >> 6, c = i & 63;
    int row = row0 + r;
    if (row < N) XO[(long)row * HID + c] = (sh[i] - smu[r]) * srs[r] * lnw[c] + lnb[c];
  }
}

// ------------------------------------------------- per-graph pooling
__global__ void pool_kernel(const float* __restrict__ x, const int* __restrict__ batch,
                            float* __restrict__ pooled, float* __restrict__ cnt, int N) {
  long t = (long)blockIdx.x * blockDim.x + threadIdx.x;
  if (t >= (long)N * HID) return;
  long n = t >> 6;
  int c = (int)(t & 63);
  int g = batch[n];
  atomicAdd(&pooled[(long)g * HID + c], x[t]);
  if (c == 0) atomicAdd(&cnt[g], 1.0f);
}

// ------------------------------------------------- readout MLP (64 graphs, trivial)
__global__ void readout_kernel(const float* __restrict__ pooled, const float* __restrict__ cnt,
                               const float* __restrict__ gf,
                               const float* __restrict__ rW1, const float* __restrict__ rb1,
                               const float* __restrict__ rln1w, const float* __restrict__ rln1b,
                               const float* __restrict__ rW2, const float* __restrict__ rb2,
                               const float* __restrict__ rln2w, const float* __restrict__ rln2b,
                               const float* __restrict__ rW3, const float* __restrict__ rb3,
                               float* __restrict__ out, int G) {
  int g = blockIdx.x * blockDim.x + threadIdx.x;
  if (g >= G) return;
  float in[HID + GLOB_D];
  float invc = 1.0f / fmaxf(cnt[g], 1.0f);
  for (int i = 0; i < HID; ++i) in[i] = pooled[(long)g * HID + i] * invc;
  for (int i = 0; i < GLOB_D; ++i) in[HID + i] = gf[g * GLOB_D + i];

  float h1[HID];
  for (int o = 0; o < HID; ++o) {
    float a = rb1[o];
    for (int k = 0; k < HID + GLOB_D; ++k) a += in[k] * rW1[k * HID + o];
    h1[o] = a;
  }
  float mu = 0.f;
  for (int i = 0; i < HID; ++i) mu += h1[i];
  mu *= (1.0f / HID);
  float var = 0.f;
  for (int i = 0; i < HID; ++i) { float d = h1[i] - mu; var += d * d; }
  var *= (1.0f / HID);
  float rs = rsqrtf(var + EPS);
  for (int i = 0; i < HID; ++i) h1[i] = fmaxf((h1[i] - mu) * rs * rln1w[i] + rln1b[i], 0.f);

  float h2[HID / 2];
  for (int o = 0; o < HID / 2; ++o) {
    float a = rb2[o];
    for (int k = 0; k < HID; ++k) a += h1[k] * rW2[k * (HID / 2) + o];
    h2[o] = a;
  }
  mu = 0.f;
  for (int i = 0; i < HID / 2; ++i) mu += h2[i];
  mu *= (2.0f / HID);
  var = 0.f;
  for (int i = 0; i < HID / 2; ++i) { float d = h2[i] - mu; var += d * d; }
  var *= (2.0f / HID);
  rs = rsqrtf(var + EPS);
  float acc = rb3[0];
  for (int i = 0; i < HID / 2; ++i)
    acc += fmaxf((h2[i] - mu) * rs * rln2w[i] + rln2b[i], 0.f) * rW3[i];
  out[g] = acc;
}

// ----------------------------------------------------------------- launcher
extern "C" void kernel_launch(void* const* d_in, const int* in_sizes, int n_in,
                              void* d_out, int out_size, void* d_ws, size_t ws_size,
                              hipStream_t stream) {
  const float* x    = (const float*)d_in[0];
  const float* ea   = (const float*)d_in[1];
  const float* gf   = (const float*)d_in[2];
  const int*   eidx = (const int*)d_in[3];
  const int*   batch= (const int*)d_in[4];
  const float* W1   = (const float*)d_in[5];
  const float* b1   = (const float*)d_in[6];
  const float* W2   = (const float*)d_in[7];
  const float* b2   = (const float*)d_in[8];
  const float* Wu   = (const float*)d_in[9];
  const float* bu   = (const float*)d_in[10];
  const float* lnw  = (const float*)d_in[11];
  const float* lnb  = (const float*)d_in[12];
  const float* rW1  = (const float*)d_in[13];
  const float* rb1  = (const float*)d_in[14];
  const float* rln1w= (const float*)d_in[15];
  const float* rln1b= (const float*)d_in[16];
  const float* rW2  = (const float*)d_in[17];
  const float* rb2  = (const float*)d_in[18];
  const float* rln2w= (const float*)d_in[19];
  const float* rln2b= (const float*)d_in[20];
  const float* rW3  = (const float*)d_in[21];
  const float* rb3  = (const float*)d_in[22];
  float* out = (float*)d_out;

  long N = in_sizes[0] / HID;
  long E = in_sizes[1] / EDGE_D;
  int  G = in_sizes[2] / GLOB_D;
  const int* src = eidx;
  const int* dst = eidx + E;

  float* ws     = (float*)d_ws;
  float* x_cur  = ws;
  float* p      = x_cur + N * HID;
  float* aggr   = p + N * HID;
  float* EAsum  = aggr + N * HID;
  float* deg    = EAsum + N * EDGE_D;   // EAsum..deg contiguous: N*(16+1) floats
  float* M1     = deg + N;
  float* bm     = M1 + HID * HID;
  float* gu     = bm + HID;
  float* pooled = gu + (long)G * HID;
  float* cnt    = pooled + (long)G * HID;  // pooled..cnt contiguous: G*(64+1)

  const int B = 256;
  // init (every call; fully deterministic, no residual state)
  copy_f32<<<(int)((N * HID + B - 1) / B), B, 0, stream>>>(x_cur, x, N * HID);
  zero_f32<<<(int)((N * (EDGE_D + 1) + B - 1) / B), B, 0, stream>>>(EAsum, N * (EDGE_D + 1));
  zero_f32<<<(int)(((long)G * (HID + 1) + B - 1) / B), B, 0, stream>>>(pooled, (long)G * (HID + 1));
  edge_presum<<<(int)((E * EDGE_D + B - 1) / B), B, 0, stream>>>(ea, dst, EAsum, deg, E);

  int nblk16 = (int)((N + 15) / 16);
  for (int i = 0; i < N_LAYERS; ++i) {
    const float* W1i = W1 + (long)i * HID * HID;
    const float* b1i = b1 + i * HID;
    const float* W2i = W2 + (long)i * (HID + EDGE_D) * HID;
    const float* b2i = b2 + i * HID;
    const float* Wui = Wu + (long)i * (HID + GLOB_D) * HID;
    const float* bui = bu + i * HID;
    fold_weights<<<1, 256, 0, stream>>>(W1i, b1i, W2i, b2i, Wui, bui, gf, G, M1, bm, gu);
    gemm_n64_wmma<<<nblk16, 128, 0, stream>>>(x_cur, M1, p, (int)N);
    init_aggr<<<(int)((N * HID + B - 1) / B), B, 0, stream>>>(
        EAsum, deg, W2i + HID * HID /*W2[64:]*/, bm, aggr, (int)N);
    scatter_edges<<<(int)((E * HID + B - 1) / B), B, 0, stream>>>(p, src, dst, aggr, E);
    update_ln_wmma<<<nblk16, 128, 0, stream>>>(aggr, Wui /*rows 0..63*/, gu, batch,
                                               lnw + i * HID, lnb + i * HID, x_cur, (int)N);
  }
  pool_kernel<<<(int)((N * HID + B - 1) / B), B, 0, stream>>>(x_cur, batch, pooled, cnt, (int)N);
  readout_kernel<<<1, 64, 0, stream>>>(pooled, cnt, gf, rW1, rb1, rln1w, rln1b,
                                       rW2, rb2, rln2w, rln2b, rW3, rb3, out, G);
}